// GroupSparseActivation_16527034155126
// MI455X (gfx1250) — compile-verified
//
#include <hip/hip_runtime.h>
#include <hip/hip_bf16.h>
#include <stdint.h>

// Problem constants (match reference)
#define B_  32
#define S_  2048
#define C_  1024
#define G_  16
#define GS_ 64
#define K_  256
#define EPS_ 1e-9f

#define CHUNK_FLOATS 128                       // 32 lanes * 4 floats (b128 each)
#define CHUNKS_PER_ROW (C_ / CHUNK_FLOATS)     // 8
#define NCHUNK (B_ * S_ * CHUNKS_PER_ROW)      // 524288
#define PIPE 4                                 // async pipeline depth per wave

typedef float vf4 __attribute__((ext_vector_type(4)));

// ---- CDNA5 async-counter wait --------------------------------------------
#if defined(__has_builtin)
#  if __has_builtin(__builtin_amdgcn_s_wait_asynccnt)
#    define WAIT_ASYNC(n) __builtin_amdgcn_s_wait_asynccnt(n)
#  endif
#endif
#ifndef WAIT_ASYNC
#  define WAIT_ASYNC(n) asm volatile("s_wait_asynccnt %0" :: "i"(n) : "memory")
#endif

// Async global -> LDS, 16 bytes per lane. vdst = LDS byte offset (low 32 bits
// of the flat shared pointer ARE the workgroup-relative LDS address on CDNA5),
// vaddr = 64-bit global address pair, GV mode (saddr = off). Uses ASYNCcnt.
__device__ __forceinline__ void async_load_b128(uint32_t lds_off, const void* gaddr) {
    asm volatile("global_load_async_to_lds_b128 %0, %1, off"
                 :: "v"(lds_off), "v"(gaddr)
                 : "memory");
}

// ---------------------------------------------------------------------------
// Pass 1: group-wise sum-of-squares -> norm bits, layout nrm[(b*G+g)*S + s].
// Each wave rings PIPE 512B chunks of x through LDS with async loads.
// ---------------------------------------------------------------------------
__global__ void gsa_norms_kernel(const float* __restrict__ x,
                                 uint32_t* __restrict__ nrm) {
    __shared__ float lds[PIPE][8][CHUNK_FLOATS];   // [buf][wave][128 floats] = 16 KB

    const int lane   = threadIdx.x & 31;
    const int wv     = threadIdx.x >> 5;
    const int nWaves = (gridDim.x * blockDim.x) >> 5;
    const int gw     = (blockIdx.x * blockDim.x + threadIdx.x) >> 5;

    auto issue = [&](int chunk, int buf) {
        int ch = chunk < NCHUNK ? chunk : (NCHUNK - 1);   // clamp dummy issues
        const float* src = x + (size_t)ch * CHUNK_FLOATS + lane * 4;
        uint32_t lds_off = (uint32_t)(uintptr_t)(&lds[buf][wv][lane * 4]);
        async_load_b128(lds_off, (const void*)src);
    };

    if (gw >= NCHUNK) return;

    #pragma unroll
    for (int j = 0; j < PIPE; ++j) issue(gw + j * nWaves, j);

    int p = 0;
    for (int chunk = gw; chunk < NCHUNK; chunk += nWaves) {
        WAIT_ASYNC(PIPE - 1);                 // oldest (current) chunk landed

        const vf4 v = *(const vf4*)&lds[p][wv][lane * 4];
        float ss = v.x * v.x + v.y * v.y + v.z * v.z + v.w * v.w;
        // reduce over each 16-lane half (one group of 64 channels per half)
        ss += __shfl_xor(ss, 1, 32);
        ss += __shfl_xor(ss, 2, 32);
        ss += __shfl_xor(ss, 4, 32);
        ss += __shfl_xor(ss, 8, 32);

        if ((lane & 15) == 0) {
            const int row = chunk >> 3;            // b*S + s
            const int gp  = chunk & 7;             // pair-of-groups index
            const int b   = row >> 11;             // /S (2048)
            const int s   = row & (S_ - 1);
            const int grp = gp * 2 + (lane >> 4);
            const float nv = sqrtf(ss + EPS_);     // monotonic; matches reference
            nrm[(((size_t)b * G_ + grp) << 11) + s] = __float_as_uint(nv);
        }
        issue(chunk + PIPE * nWaves, p);           // refill this ring slot
        p = (p + 1) & (PIPE - 1);
    }
    // up to PIPE dangling (clamped) async ops remain; s_endpgm waits idle.
}

// ---------------------------------------------------------------------------
// Pass 2: per (b,g) row of 2048 norm-bit values, exact top-K via MSB-first
// 8-bit radix select; ties kept by ascending seq index (stable, like top_k).
// Overwrites the row in place with a 1.0f / 0.0f mask.
// 512 blocks x 256 threads; each thread owns 8 consecutive elements.
// ---------------------------------------------------------------------------
__global__ void gsa_topk_mask_kernel(uint32_t* __restrict__ nm) {
    __shared__ uint32_t hist[256];
    __shared__ uint32_t tsum[256];
    __shared__ uint32_t s_prefix, s_want;

    const int t = threadIdx.x;
    uint32_t* rowp = nm + (size_t)blockIdx.x * S_;

    // each thread loads its 8 contiguous values (index order preserved)
    uint32_t myv[8];
    {
        const uint4 a = ((const uint4*)rowp)[2 * t + 0];
        const uint4 b = ((const uint4*)rowp)[2 * t + 1];
        myv[0] = a.x; myv[1] = a.y; myv[2] = a.z; myv[3] = a.w;
        myv[4] = b.x; myv[5] = b.y; myv[6] = b.z; myv[7] = b.w;
    }
    if (t == 0) { s_prefix = 0u; s_want = K_; }
    __syncthreads();

    // 4 radix passes, MSB first (positive floats: bit order == value order)
    for (int p = 3; p >= 0; --p) {
        const int shift = p * 8;
        hist[t] = 0u;
        __syncthreads();
        const uint32_t pref = s_prefix;
        const uint32_t hm = (p == 3) ? 0u : (0xFFFFFFFFu << (shift + 8));
        #pragma unroll
        for (int i = 0; i < 8; ++i) {
            const uint32_t v = myv[i];
            if ((v & hm) == (pref & hm))
                atomicAdd(&hist[(v >> shift) & 255u], 1u);
        }
        __syncthreads();
        if (t == 0) {
            uint32_t want = s_want, acc = 0u;
            int d = 255;
            for (; d > 0; --d) {
                const uint32_t c = hist[d];
                if (acc + c >= want) break;
                acc += c;
            }
            s_prefix = pref | ((uint32_t)d << shift);
            s_want   = want - acc;           // rank within the tied digit class
        }
        __syncthreads();
    }

    const uint32_t T   = s_prefix;   // exact K-th largest value (bits)
    const uint32_t rem = s_want;     // how many ties-at-T to keep (lowest s first)

    // stable rank of tied elements: block exclusive scan of per-thread tie counts
    uint32_t eqp[8];
    uint32_t tot = 0u;
    #pragma unroll
    for (int i = 0; i < 8; ++i) { eqp[i] = tot; tot += (myv[i] == T) ? 1u : 0u; }
    tsum[t] = tot;
    __syncthreads();
    for (int off = 1; off < 256; off <<= 1) {
        const uint32_t add = (t >= off) ? tsum[t - off] : 0u;
        __syncthreads();
        tsum[t] += add;
        __syncthreads();
    }
    const uint32_t excl = tsum[t] - tot;

    float m[8];
    #pragma unroll
    for (int i = 0; i < 8; ++i) {
        const uint32_t v = myv[i];
        const bool keep = (v > T) || ((v == T) && ((excl + eqp[i]) < rem));
        m[i] = keep ? 1.0f : 0.0f;
    }
    // write mask in place (each thread owns its 8 slots)
    float4* outp = (float4*)(rowp + t * 8);
    outp[0] = make_float4(m[0], m[1], m[2], m[3]);
    outp[1] = make_float4(m[4], m[5], m[6], m[7]);
}

// ---------------------------------------------------------------------------
// Pass 3: out = x * mask[b, g, s]  (vf4 streaming, grid-stride).
// x is on its last read and out is never re-read -> NT temporal hints so the
// 256 MB output stream does not evict pass-1-warmed x lines from the 192 MB L2.
// ---------------------------------------------------------------------------
__global__ void gsa_apply_kernel(const float* __restrict__ x,
                                 const float* __restrict__ mask,
                                 float* __restrict__ out) {
    const size_t nf4    = (size_t)B_ * S_ * C_ / 4;   // 16,777,216
    const size_t stride = (size_t)gridDim.x * blockDim.x;
    const vf4* __restrict__ xv = (const vf4*)x;
    vf4* __restrict__ ov       = (vf4*)out;
    for (size_t i = (size_t)blockIdx.x * blockDim.x + threadIdx.x;
         i < nf4; i += stride) {
        vf4 v = __builtin_nontemporal_load(xv + i);
        const size_t f   = i * 4;
        const int    c   = (int)(f & (C_ - 1));
        const size_t row = f >> 10;                 // b*S + s
        const int    b   = (int)(row >> 11);
        const int    s   = (int)(row & (S_ - 1));
        const int    g   = c >> 6;
        const float  mm  = mask[(((size_t)b * G_ + g) << 11) + s];
        v *= mm;
        __builtin_nontemporal_store(v, ov + i);
    }
}

extern "C" void kernel_launch(void* const* d_in, const int* in_sizes, int n_in,
                              void* d_out, int out_size, void* d_ws, size_t ws_size,
                              hipStream_t stream) {
    const float* x   = (const float*)d_in[0];
    float*       out = (float*)d_out;
    uint32_t*    nrm = (uint32_t*)d_ws;     // B*G*S uint32 = 4 MB, then mask in place

    gsa_norms_kernel<<<4096, 256, 0, stream>>>(x, nrm);
    gsa_topk_mask_kernel<<<B_ * G_, 256, 0, stream>>>(nrm);
    gsa_apply_kernel<<<8192, 256, 0, stream>>>(x, (const float*)d_ws, out);
}